// Conv2Central_52261162058495
// MI455X (gfx1250) — compile-verified
//
#include <hip/hip_runtime.h>

// Fused stencil^4 (batch flips cancel) as a separable 5x5 filter,
// vertical pass executed on the f32 WMMA pipe (V_WMMA_F32_16X16X4_F32).

typedef __attribute__((ext_vector_type(2))) float v2f;
typedef __attribute__((ext_vector_type(8))) float v8f;

#define NN 32
#define HH 1024
#define WW 1024

// composed 1D taps: [1,0.5] convolved with itself 4x
__device__ __forceinline__ float ctap(int t) {
    float v = 0.0f;
    v = (t == 0) ? 1.0f    : v;
    v = (t == 1) ? 2.0f    : v;
    v = (t == 2) ? 1.5f    : v;
    v = (t == 3) ? 0.5f    : v;
    v = (t == 4) ? 0.0625f : v;
    return v;
}

__global__ __launch_bounds__(256) void stencil4_wmma(const float* __restrict__ img,
                                                     float* __restrict__ out) {
    const int lane      = threadIdx.x & 31;
    const int waveInBlk = threadIdx.x >> 5;        // wave32: 8 waves per 256-thread block
    const int wave      = blockIdx.x * 8 + waveInBlk;

    // 64x64 tiles of 16x16 per image, 32 images -> 131072 waves total
    const int img_n = wave >> 12;
    const int t     = wave & 4095;
    const int row0  = (t >> 6) << 4;
    const int col0  = (t & 63) << 4;

    const int n    = lane & 15;   // N column (B/D) and A row (M) for this lane
    const int half = lane >> 4;   // which K-half of A/B this lane supplies
    const int coln = col0 + n;

    const unsigned ibase = (unsigned)img_n * (HH * WW);

    // Horizontal 5-tap pass. Column-OOB (only possible on the last tile column)
    // is folded into the coefficients; addresses are clamped so loads are
    // unconditional (no exec divergence), zeroing happens via cndmask.
    const float C0 = 1.0f;                                   // coln < W always
    const float C1 = (coln + 1 < WW) ? 2.0f    : 0.0f;
    const float C2 = (coln + 2 < WW) ? 1.5f    : 0.0f;
    const float C3 = (coln + 3 < WW) ? 0.5f    : 0.0f;
    const float C4 = (coln + 4 < WW) ? 0.0625f : 0.0f;
    const int cc1 = (coln + 1 < WW) ? coln + 1 : WW - 1;
    const int cc2 = (coln + 2 < WW) ? coln + 2 : WW - 1;
    const int cc3 = (coln + 3 < WW) ? coln + 3 : WW - 1;
    const int cc4 = (coln + 4 < WW) ? coln + 4 : WW - 1;

    auto yh = [&](int r) -> float {
        const int  gr  = row0 + r;
        const bool rok = gr < HH;
        const int  grc = rok ? gr : (HH - 1);
        const float* p = img + ibase + (unsigned)grc * WW;
        float a = C0 * p[coln] + C1 * p[cc1] + C2 * p[cc2] + C3 * p[cc3] + C4 * p[cc4];
        return rok ? a : 0.0f;   // row-OOB -> zero-extension
    };

    // B operand (Yh, 20x16 split into five 4x16 chunks).
    // f32 B 4x16 layout (mirror of ISA A 16x4): VGPR p, lanes 0-15 hold row
    // K=4k+p (N=lane), lanes 16-31 hold row K=4k+2+p.
    v2f B[5];
#pragma unroll
    for (int k = 0; k < 5; ++k) {
        B[k].x = yh(4 * k + 2 * half + 0);
        B[k].y = yh(4 * k + 2 * half + 1);
    }

    // A operand: banded 16x20 matrix M[m,col] = c[col-m], K-chunked.
    // ISA layout: lanes 0-15 rows M=0..15 with K={0,1} in VGPR0/1,
    // lanes 16-31 same rows with K={2,3}.
    v8f acc = {};
#pragma unroll
    for (int k = 0; k < 5; ++k) {
        const int t0 = 4 * k + 2 * half - n;   // c-index for this lane's K slot
        v2f a;
        a.x = ctap(t0);
        a.y = ctap(t0 + 1);
        // 8 args: (neg_a, A, neg_b, B, c_mod, C, reuse_a, reuse_b)
        acc = __builtin_amdgcn_wmma_f32_16x16x4_f32(
            false, a, false, B[k], (short)0, acc, false, false);
    }

    // D layout: VGPR r -> row M=r (lanes 0-15) / M=r+8 (lanes 16-31), N=lane&15.
    const unsigned obase = ibase + (unsigned)(row0 + 8 * half) * WW + coln;
#pragma unroll
    for (int r = 0; r < 8; ++r)
        out[obase + (unsigned)r * WW] = acc[r];
}

extern "C" void kernel_launch(void* const* d_in, const int* in_sizes, int n_in,
                              void* d_out, int out_size, void* d_ws, size_t ws_size,
                              hipStream_t stream) {
    (void)in_sizes; (void)n_in; (void)d_ws; (void)ws_size; (void)out_size;
    const float* img = (const float*)d_in[0];
    float*       out = (float*)d_out;

    const int waves  = NN * 64 * 64;       // 131072 tiles, one wave each
    const int blocks = waves / 8;          // 8 wave32s per 256-thread block
    stencil4_wmma<<<blocks, 256, 0, stream>>>(img, out);
}